// LaplacianLoss_62929860821305
// MI455X (gfx1250) — compile-verified
//
#include <hip/hip_runtime.h>

#define N_DIM 16384
#define M_DIM 8192
#define KSPLIT 32
#define KCHUNK (N_DIM / KSPLIT)   // 512
#define TILE_M 16

typedef __attribute__((ext_vector_type(2))) float v2f;
typedef __attribute__((ext_vector_type(8))) float v8f;

// ---------------------------------------------------------------------------
// Kernel 1: zero the per-row accumulator (M x 3) and the scalar output.
// ---------------------------------------------------------------------------
__global__ __launch_bounds__(256) void lap_init_kernel(float* __restrict__ rowacc,
                                                       float* __restrict__ out) {
  int tid = blockIdx.x * blockDim.x + threadIdx.x;
  if (tid == 0) out[0] = 0.0f;
  for (int i = tid; i < M_DIM * 3; i += gridDim.x * blockDim.x) rowacc[i] = 0.0f;
}

// ---------------------------------------------------------------------------
// Kernel 2: masked-row GEMM via V_WMMA_F32_16X16X4_F32.
//   grid.x = M/16 row tiles, grid.y = KSPLIT K-slices, 1 wave per block.
//   A (16x4 f32): lanes 0-15 hold K=k,k+1 ; lanes 16-31 hold K=k+2,k+3.
//   B (4x16 f32): lane = N column, same K half-split. Columns 3..15 are zero;
//     we stage V' into LDS padded to 4 columns (col 3 = 0) and clamp col,
//     so every lane issues an unconditional ds_load_b64 (no EXEC dance).
//   LDS layout: sB2[(k>>1)*8 + c*2 + (k&1)] = V'[k0+k][c]  (c==3 -> 0)
//     => the (k, k+1) pair a lane needs per WMMA is one aligned b64.
// ---------------------------------------------------------------------------
__global__ __launch_bounds__(32) void lap_gemm_kernel(const float* __restrict__ L,
                                                      const float* __restrict__ Vp,
                                                      const int* __restrict__ mask,
                                                      float* __restrict__ rowacc) {
  __shared__ __align__(16) float sB2[KCHUNK * 4];   // 8 KB padded V' slice

  const int lane = threadIdx.x;      // 0..31 (wave32)
  const int tile = blockIdx.x;       // 0..511
  const int k0   = blockIdx.y * KCHUNK;

  // Stage + interleave V'[k0 : k0+KCHUNK, 0:3] into LDS, zero-padding col 3.
  for (int i = lane; i < KCHUNK * 4; i += 32) {
    const int k = i >> 2;            // 0..KCHUNK-1
    const int c = i & 3;             // 0..3
    const float v = (c < 3) ? Vp[(k0 + k) * 3 + c] : 0.0f;
    sB2[(k >> 1) * 8 + c * 2 + (k & 1)] = v;
  }
  __syncthreads();

  const int  col  = lane & 15;       // N column for B/C/D, M row for A
  const int  half = lane >> 4;       // K half-select
  const int  c4   = (col < 3) ? col : 3;           // clamp into padded columns
  const int  row  = mask[tile * TILE_M + col];     // gathered L row for this lane
  const float* Arow  = L + (size_t)row * N_DIM + k0 + half * 2;
  const float* Bbase = &sB2[half * 8 + c4 * 2];

  v8f acc = {};
  #pragma unroll 8
  for (int kk = 0; kk < KCHUNK; kk += 4) {
    if ((kk & 15) == 0)
      __builtin_prefetch(Arow + kk + 256, 0, 1);   // global_prefetch_b8, 1KB ahead
    v2f a = *(const v2f*)(Arow + kk);              // K = k0+kk+2*half, +1
    v2f b = *(const v2f*)(Bbase + kk * 4);         // same two K rows, column c4
    // (neg_a, A, neg_b, B, c_mod, C, reuse_a, reuse_b)
    acc = __builtin_amdgcn_wmma_f32_16x16x4_f32(false, a, false, b,
                                                (short)0, acc, false, false);
  }

  // Scatter partial dot products: VGPR j holds row (tile*16 + half*8 + j), col.
  if (col < 3) {
    const int mbase = tile * TILE_M + half * 8;
    #pragma unroll
    for (int j = 0; j < 8; ++j) {
      __hip_atomic_fetch_add(&rowacc[(mbase + j) * 3 + col], acc[j],
                             __ATOMIC_RELAXED, __HIP_MEMORY_SCOPE_AGENT);
    }
  }
}

// ---------------------------------------------------------------------------
// Kernel 3: loss = sum over masked rows of (rowacc - delta_V[mask])^2.
// ---------------------------------------------------------------------------
__global__ __launch_bounds__(256) void lap_loss_kernel(const float* __restrict__ rowacc,
                                                       const float* __restrict__ dV,
                                                       const int* __restrict__ mask,
                                                       float* __restrict__ out) {
  int tid = blockIdx.x * blockDim.x + threadIdx.x;
  float local = 0.0f;
  for (int i = tid; i < M_DIM * 3; i += gridDim.x * blockDim.x) {
    int m = i / 3, c = i % 3;
    float d = rowacc[i] - dV[(size_t)mask[m] * 3 + c];
    local += d * d;
  }
  // wave32 tree reduction
  #pragma unroll
  for (int off = 16; off > 0; off >>= 1) local += __shfl_down(local, off, 32);
  if ((threadIdx.x & 31) == 0) {
    __hip_atomic_fetch_add(out, local, __ATOMIC_RELAXED, __HIP_MEMORY_SCOPE_AGENT);
  }
}

// ---------------------------------------------------------------------------
// Launch. Inputs: [0]=V_prime (N*3 f32), [1]=laplacian (N*N f32),
//                 [2]=delta_V (N*3 f32), [3]=mask (M int).
// ---------------------------------------------------------------------------
extern "C" void kernel_launch(void* const* d_in, const int* in_sizes, int n_in,
                              void* d_out, int out_size, void* d_ws, size_t ws_size,
                              hipStream_t stream) {
  const float* Vp   = (const float*)d_in[0];
  const float* L    = (const float*)d_in[1];
  const float* dV   = (const float*)d_in[2];
  const int*   mask = (const int*)d_in[3];
  float* out    = (float*)d_out;
  float* rowacc = (float*)d_ws;   // M_DIM * 3 floats = 96 KB

  lap_init_kernel<<<96, 256, 0, stream>>>(rowacc, out);

  dim3 grid(M_DIM / TILE_M, KSPLIT);   // 512 x 32 blocks, 1 wave each
  lap_gemm_kernel<<<grid, 32, 0, stream>>>(L, Vp, mask, rowacc);

  lap_loss_kernel<<<64, 256, 0, stream>>>(rowacc, dV, mask, out);
}